// ModelTorch_65335042507146
// MI455X (gfx1250) — compile-verified
//
#include <hip/hip_runtime.h>

typedef __attribute__((ext_vector_type(2))) float v2f;
typedef __attribute__((ext_vector_type(8))) float v8f;

#define DD    128
#define LL    1024
#define SS    8192
#define HKV   8
#define NWAVE 8
#define NTILE (LL / 16)   // 64 tiles of 16 rows

__global__ __launch_bounds__(256)
void sparse_attn_decode(const float* __restrict__ q,
                        const float* __restrict__ Kt,
                        const float* __restrict__ Vt,
                        const int*   __restrict__ ind,
                        const int*   __restrict__ nnz,
                        const int*   __restrict__ gqa,
                        float*       __restrict__ out)
{
    __shared__ float s_q[DD];
    __shared__ float s_sc[LL];          // scores -> unnormalized probs
    __shared__ float s_red[NWAVE];
    __shared__ float s_part[NWAVE * DD];
    __shared__ float s_stat[2];         // [0]=max, [1]=1/sum

    const int bh   = blockIdx.x;        // b*32 + h
    const int h    = bh & 31;
    const int b    = bh >> 5;
    const int tid  = threadIdx.x;
    const int wave = tid >> 5;
    const int lane = tid & 31;
    const int ln   = lane & 15;         // column / N index
    const int hi   = lane >> 4;         // lane-half -> K offset 2*hi

    const int g   = gqa[0];
    const int kvh = h / g;

    int n = nnz[bh];
    if (n < 1) n = 1;

    if (tid < DD) s_q[tid] = q[(long long)bh * DD + tid];

    const long long kvbase = ((long long)b * HKV + kvh) * (long long)SS * DD;
    const float* Kb = Kt + kvbase;
    const float* Vb = Vt + kvbase;
    const int*   ib = ind + (long long)bh * LL;
    __syncthreads();

    const float scale = 0.08838834764831845f;   // 1/sqrt(128)

    // ---------------- Phase 1: scores = (q . K_gathered) * scale ----------------
    for (int t = wave; t < NTILE; t += NWAVE) {
        const int l0 = t * 16;
        if (l0 >= n) {                    // whole tile masked: no gather needed
            if (lane < 16) s_sc[l0 + lane] = -1e30f;
            continue;
        }
        const int row = l0 + ln;
        const int idx = (row < n) ? ib[row] : 0;
        const float* kr = Kb + (long long)idx * DD + 2 * hi;

        v8f c = {};
        #pragma unroll
        for (int d = 0; d < DD; d += 4) {
            // A (16x4 f32): row 0 = q[d..d+3]; all rows replicated (only row 0 of D used)
            v2f a;
            a.x = s_q[d + 2 * hi];
            a.y = s_q[d + 2 * hi + 1];
            // B (4x16 f32): B[k][n] = K_row_n[d+k]
            v2f bm = *(const v2f*)(kr + d);
            c = __builtin_amdgcn_wmma_f32_16x16x4_f32(false, a, false, bm,
                                                      (short)0, c, false, false);
        }
        float sc = c[0] * scale;          // score for column ln (all rows identical)
        if (row >= n) sc = -1e30f;
        if (lane < 16) s_sc[row] = sc;
    }
    __syncthreads();

    // ---------------- Block softmax statistics ----------------
    float m = -1e30f;
    for (int i = tid; i < LL; i += 256) m = fmaxf(m, s_sc[i]);
    #pragma unroll
    for (int o = 16; o > 0; o >>= 1) m = fmaxf(m, __shfl_xor(m, o, 32));
    if (lane == 0) s_red[wave] = m;
    __syncthreads();
    if (wave == 0) {
        float mm = (lane < NWAVE) ? s_red[lane] : -1e30f;
        #pragma unroll
        for (int o = 16; o > 0; o >>= 1) mm = fmaxf(mm, __shfl_xor(mm, o, 32));
        if (lane == 0) s_stat[0] = mm;
    }
    __syncthreads();
    m = s_stat[0];

    float sum = 0.f;
    for (int i = tid; i < LL; i += 256) {
        float e = __expf(s_sc[i] - m);    // masked rows (-1e30) -> 0
        s_sc[i] = e;                      // keep unnormalized probs in LDS
        sum += e;
    }
    #pragma unroll
    for (int o = 16; o > 0; o >>= 1) sum += __shfl_xor(sum, o, 32);
    if (lane == 0) s_red[wave] = sum;
    __syncthreads();
    if (wave == 0) {
        float ss = (lane < NWAVE) ? s_red[lane] : 0.f;
        #pragma unroll
        for (int o = 16; o > 0; o >>= 1) ss += __shfl_xor(ss, o, 32);
        if (lane == 0) s_stat[1] = 1.0f / ss;
    }
    __syncthreads();
    const float inv = s_stat[1];

    // ---------------- Phase 2: out = probs . V_gathered ----------------
    v8f acc[8];
    #pragma unroll
    for (int cc = 0; cc < 8; ++cc) acc[cc] = (v8f){};

    for (int t = wave; t < NTILE; t += NWAVE) {
        const int l0 = t * 16;
        if (l0 >= n) continue;            // probs are exactly zero
        #pragma unroll
        for (int j = 0; j < 4; ++j) {
            const int lj = l0 + 4 * j + 2 * hi;
            // A (16x4 f32): row 0 = p[l0+4j .. l0+4j+3]; replicated rows
            v2f a;
            a.x = s_sc[lj];
            a.y = s_sc[lj + 1];
            const int i0 = ib[lj];
            const int i1 = ib[lj + 1];
            const float* v0p = Vb + (long long)i0 * DD + ln;
            const float* v1p = Vb + (long long)i1 * DD + ln;
            #pragma unroll
            for (int cc = 0; cc < 8; ++cc) {
                // B (4x16 f32): B[k][n] = V_row(l0+4j+k)[16*cc + n]
                v2f bm;
                bm.x = v0p[cc * 16];
                bm.y = v1p[cc * 16];
                acc[cc] = __builtin_amdgcn_wmma_f32_16x16x4_f32(false, a, false, bm,
                                                                (short)0, acc[cc],
                                                                false, false);
            }
        }
    }

    // partial output of this wave: D row 0, lanes 0..15 per 16-wide chunk
    if (lane < 16) {
        #pragma unroll
        for (int cc = 0; cc < 8; ++cc)
            s_part[wave * DD + cc * 16 + lane] = acc[cc][0];
    }
    __syncthreads();

    if (tid < DD) {
        float o = 0.f;
        #pragma unroll
        for (int w = 0; w < NWAVE; ++w) o += s_part[w * DD + tid];
        out[(long long)bh * DD + tid] = o * inv;
    }
}

extern "C" void kernel_launch(void* const* d_in, const int* in_sizes, int n_in,
                              void* d_out, int out_size, void* d_ws, size_t ws_size,
                              hipStream_t stream) {
    const float* q   = (const float*)d_in[0];
    const float* K   = (const float*)d_in[1];
    const float* V   = (const float*)d_in[2];
    const int*   ind = (const int*)d_in[3];
    const int*   nnz = (const int*)d_in[4];
    const int*   gqa = (const int*)d_in[5];
    float*       out = (float*)d_out;

    dim3 grid(16 * 32);   // one block per (b, h)
    sparse_attn_decode<<<grid, 256, 0, stream>>>(q, K, V, ind, nnz, gqa, out);
}